// PreprocessingLayer_23192823398835
// MI455X (gfx1250) — compile-verified
//
#include <hip/hip_runtime.h>
#include <math.h>
#include <stdint.h>

#define T_FRAMES 384
#define N_LM     62
#define N_PTS    543
#define B_BATCH  256
#define FEAT     744
#define EPSF     1e-8f
#define TS       16   // frames per block in pass 2

// 20 face + 21 left-hand (501..521) + 21 right-hand (522..542)
__constant__ int c_lm[N_LM] = {
    33, 133, 362, 263, 61, 291, 199, 419, 17, 84, 17, 314, 405, 320, 307, 375, 321, 308, 324, 318,
    501, 502, 503, 504, 505, 506, 507, 508, 509, 510, 511, 512, 513, 514, 515, 516, 517, 518, 519, 520, 521,
    522, 523, 524, 525, 526, 527, 528, 529, 530, 531, 532, 533, 534, 535, 536, 537, 538, 539, 540, 541, 542
};

__device__ __forceinline__ float nan0(float v) { return (v != v) ? 0.0f : v; }

// ---------------- Pass 1: per-batch nose mean + inv(std+eps) ----------------
__global__ __launch_bounds__(256)
void pre_stats_kernel(const float* __restrict__ x, float* __restrict__ stats)
{
    const int b = blockIdx.x;
    const size_t base = (size_t)b * T_FRAMES * N_PTS * 3;

    float s0 = 0.f, s1 = 0.f, q0 = 0.f, q1 = 0.f, ns0 = 0.f, ns1 = 0.f;

    // std accumulation over (T, 62 landmarks), NaN -> 0, channels 0 and 1 only
    for (int i = threadIdx.x; i < T_FRAMES * N_LM; i += 256) {
        const int t = i / N_LM;
        const int n = i - t * N_LM;
        const size_t p = base + ((size_t)t * N_PTS + c_lm[n]) * 3;
        float v0 = x[p + 0];
        float v1 = x[p + 1];
        v0 = (v0 != v0) ? 0.f : v0;
        v1 = (v1 != v1) ? 0.f : v1;
        s0 += v0; s1 += v1;
        q0 += v0 * v0; q1 += v1 * v1;
    }
    // nose mean over T, landmark 17, NaN -> 0.5
    for (int t = threadIdx.x; t < T_FRAMES; t += 256) {
        const size_t p = base + ((size_t)t * N_PTS + 17) * 3;
        float v0 = x[p + 0];
        float v1 = x[p + 1];
        ns0 += (v0 != v0) ? 0.5f : v0;
        ns1 += (v1 != v1) ? 0.5f : v1;
    }

    __shared__ float red[256];
    auto reduce = [&](float v) -> float {
        red[threadIdx.x] = v;
        __syncthreads();
        for (int off = 128; off > 0; off >>= 1) {
            if ((int)threadIdx.x < off) red[threadIdx.x] += red[threadIdx.x + off];
            __syncthreads();
        }
        float r = red[0];
        __syncthreads();
        return r;
    };

    s0 = reduce(s0); s1 = reduce(s1);
    q0 = reduce(q0); q1 = reduce(q1);
    ns0 = reduce(ns0); ns1 = reduce(ns1);

    if (threadIdx.x == 0) {
        const float n = (float)(T_FRAMES * N_LM);
        float var0 = (q0 - s0 * s0 / n) / (n - 1.f);
        float var1 = (q1 - s1 * s1 / n) / (n - 1.f);
        var0 = var0 > 0.f ? var0 : 0.f;
        var1 = var1 > 0.f ? var1 : 0.f;
        stats[b * 4 + 0] = ns0 * (1.f / (float)T_FRAMES);
        stats[b * 4 + 1] = ns1 * (1.f / (float)T_FRAMES);
        stats[b * 4 + 2] = 1.f / (sqrtf(var0) + EPSF);
        stats[b * 4 + 3] = 1.f / (sqrtf(var1) + EPSF);
    }
}

// ---------------- Pass 2: feature construction ----------------
// Block: 256 threads; handles (batch b, frames t0..t0+TS-1).
// Raw landmark xy pairs are gathered into LDS with CDNA5 async
// global->LDS copies (ASYNCcnt); normalization happens at consume time.
__global__ __launch_bounds__(256)
void pre_feat_kernel(const float* __restrict__ x,
                     const float* __restrict__ stats,
                     float* __restrict__ out)
{
    const int b  = blockIdx.y;
    const int t0 = blockIdx.x * TS;

    __shared__ float2 sraw[TS + 2][64];

    const float m0 = stats[b * 4 + 0];
    const float m1 = stats[b * 4 + 1];
    const float i0 = stats[b * 4 + 2];
    const float i1 = stats[b * 4 + 3];

    const size_t xb = (size_t)b * T_FRAMES * N_PTS * 3;

    // Zero-fill padding slots (tail frames of last chunk + landmark slots 62/63).
    for (int i = threadIdx.x; i < (TS + 2) * 64; i += 256) {
        const int f = i >> 6;
        const int n = i & 63;
        if ((t0 + f) >= T_FRAMES || n >= N_LM)
            sraw[f][n] = make_float2(0.f, 0.f);
    }

    // Async gather phase: raw (x0,x1) pairs -> LDS, no VGPR round-trip.
    // inst offset applies to both global and LDS addresses, so one address
    // pair serves both floats of the xy pair.
    for (int i = threadIdx.x; i < (TS + 2) * N_LM; i += 256) {
        const int f = i / N_LM;
        const int n = i - f * N_LM;
        const int t = t0 + f;
        if (t < T_FRAMES) {
            const size_t  p  = xb + ((size_t)t * N_PTS + c_lm[n]) * 3;
            const uint64_t g0 = (uint64_t)(uintptr_t)(x + p);
            const uint32_t l0 = (uint32_t)(uintptr_t)(&sraw[f][n]);
            asm volatile("global_load_async_to_lds_b32 %0, %1, off"
                         :: "v"(l0), "v"(g0) : "memory");
            asm volatile("global_load_async_to_lds_b32 %0, %1, off offset:4"
                         :: "v"(l0), "v"(g0) : "memory");
        }
    }
    asm volatile("s_wait_asynccnt 0x0" ::: "memory");
    __syncthreads();

    const int n  = threadIdx.x & 63;   // landmark slot
    const int fr = threadIdx.x >> 6;   // 0..3

    for (int k = 0; k < TS / 4; ++k) {
        const int f = fr + k * 4;      // 0..TS-1
        const int t = t0 + f;
        if (n < N_LM) {
            const float2 R0 = sraw[f][n];
            const float2 R1 = sraw[f + 1][n];
            const float2 R2 = sraw[f + 2][n];
            const float2 P0 = make_float2((R0.x - m0) * i0, (R0.y - m1) * i1);
            const float2 P1 = make_float2((R1.x - m0) * i0, (R1.y - m1) * i1);
            const float2 P2 = make_float2((R2.x - m0) * i0, (R2.y - m1) * i1);
            const bool v1 = (t + 1) < T_FRAMES;
            const bool v2 = (t + 2) < T_FRAMES;

            float dxx = v1 ? (P1.x - P0.x) : 0.f;
            float dxy = v1 ? (P1.y - P0.y) : 0.f;
            float d2x = v2 ? (P2.x - P0.x) : 0.f;
            float d2y = v2 ? (P2.y - P0.y) : 0.f;

            float rx = 0.f, ry = 0.f;
            if (v1 && n < (N_LM - 1)) {
                const float2 RQ = sraw[f][n + 1];
                rx = P1.x - (RQ.x - m0) * i0;
                ry = P1.y - (RQ.y - m1) * i1;
            }

            float tc = 0.f;
            if (v2) {
                const float ax = P1.x - P0.x, ay = P1.y - P0.y;
                const float bx = P2.x - P1.x, by = P2.y - P1.y;
                float na = sqrtf(ax * ax + ay * ay);
                float nb = sqrtf(bx * bx + by * by);
                na = (na < EPSF) ? EPSF : na;   // NaN-preserving max
                nb = (nb < EPSF) ? EPSF : nb;
                tc = (ax * bx + ay * by) / (na * nb);
            }

            const float mm = sqrtf(dxx * dxx + dxy * dxy);
            const float md = atan2f(dxy, dxx);   // atan2(0,0)=0 matches arctan2

            float* row = out + (size_t)(b * T_FRAMES + t) * FEAT;
            // xy:0  dx:124  dx2:248  rel:372  tc:496  mm:620  md:682
            *(float2*)(row +   0 + 2 * n) = make_float2(nan0(P0.x), nan0(P0.y));
            *(float2*)(row + 124 + 2 * n) = make_float2(nan0(dxx),  nan0(dxy));
            *(float2*)(row + 248 + 2 * n) = make_float2(nan0(d2x),  nan0(d2y));
            *(float2*)(row + 372 + 2 * n) = make_float2(nan0(rx),   nan0(ry));
            const float tcc = nan0(tc);
            *(float2*)(row + 496 + 2 * n) = make_float2(tcc, tcc);
            row[620 + n] = nan0(mm);
            row[682 + n] = nan0(md);
        }
    }
}

extern "C" void kernel_launch(void* const* d_in, const int* in_sizes, int n_in,
                              void* d_out, int out_size, void* d_ws, size_t ws_size,
                              hipStream_t stream)
{
    const float* x   = (const float*)d_in[0];
    float* out       = (float*)d_out;
    float* stats     = (float*)d_ws;   // 256 batches * 4 floats = 4 KB

    pre_stats_kernel<<<B_BATCH, 256, 0, stream>>>(x, stats);

    dim3 grid(T_FRAMES / TS, B_BATCH);
    pre_feat_kernel<<<grid, 256, 0, stream>>>(x, stats, out);
}